// VectorQuantizerEMA_14654428413994
// MI455X (gfx1250) — compile-verified
//
#include <hip/hip_runtime.h>
#include <hip/hip_bf16.h>

typedef float v2f __attribute__((ext_vector_type(2)));
typedef float v8f __attribute__((ext_vector_type(8)));

#define N_ROWS    32768
#define EDIM      64
#define NE        1024
#define N_TILES   64        // NE / 16
#define Z_ELEMS   2097152   // 32*64*32*32
#define DECAYF    0.99f
#define OMD       0.01f     // 1 - decay
#define BETAF     0.25f
#define EPSF      1e-5f

// d_out layout (floats): [loss(1) | z_q_st(2097152) | perp(1) | min_enc(33554432) | idx(32768)]
#define ZQ_OFF    1
#define PERP_OFF  2097153
#define ME_OFF    2097154
#define IDX_OFF   35651586

// ---------------- kernel 0: zero the accumulators (must run every launch) ---------------
__global__ void vq_init(float* cs, float* se, float* out) {
    int i = blockIdx.x * 256 + threadIdx.x;      // grid covers 65536
    if (i < NE)      cs[i] = 0.0f;
    if (i < NE*EDIM) se[i] = 0.0f;
    if (i == 0)      out[0] = 0.0f;              // loss accumulator
}

// ---------------- kernel 1: -0.5 * ||e||^2 (folded into the WMMA C accumulator) ---------------
__global__ void vq_enorm(const float* __restrict__ emb, float* __restrict__ en2h) {
    int e = blockIdx.x * 256 + threadIdx.x;
    if (e >= NE) return;
    const float* r = emb + (e << 6);
    float s = 0.0f;
    #pragma unroll
    for (int c = 0; c < EDIM; ++c) s += r[c] * r[c];
    en2h[e] = -0.5f * s;
}

// ---------------- kernel 2: fp32 WMMA scores + branchless argmax + cluster stats ---------------
// score[m][n] = dot(code m, zrow n) - 0.5*||code m||^2  (C initialized from en2h)
// argmin distance == argmax score. Lane owns z-row n = lane&15 and 8 code rows.
__global__ __launch_bounds__(256) void vq_argmin(
        const float* __restrict__ z, const float* __restrict__ emb,
        const float* __restrict__ en2h, int* __restrict__ idx_out,
        float* __restrict__ cluster_size, float* __restrict__ sum_enc) {
    const int lane  = threadIdx.x & 31;
    const int wv    = threadIdx.x >> 5;
    const int rt    = blockIdx.x * 8 + wv;       // row tile (2048 total)
    const int nbase = rt << 4;

    const int mlane = lane & 15;
    const int khalf = (lane >> 4) << 1;          // 0 or 2

    // zf[n][c] = z[(n>>10)*65536 + c*1024 + (n&1023)]
    const int nrow = nbase + mlane;
    const float* zrow = z + ((nrow >> 10) << 16) + (nrow & 1023);

    // B operand: 64 x 16, per lane 2 floats per k-step, 16 steps = 32 VGPRs, loaded once.
    v2f breg[16];
    #pragma unroll
    for (int s = 0; s < 16; ++s) {
        const int kb = (s << 2) + khalf;
        breg[s].x = zrow[(size_t)kb * 1024];
        breg[s].y = zrow[(size_t)(kb + 1) * 1024];
    }

    float best = -3.4e38f;                       // tracking MAX of score
    int bestIdx = 0;
    const int mofs = (lane >> 4) << 3;           // 0 or 8: M offset of this half-wave in D

    for (int t = 0; t < N_TILES; ++t) {
        const float* arow = emb + (((t << 4) + mlane) << 6) + khalf;
        if (t + 1 < N_TILES) __builtin_prefetch(arow + (16 << 6), 0, 0);

        // C init = -0.5*||e||^2 for the 8 codes this lane owns (two aligned b128 loads)
        const float* e2p = en2h + (t << 4) + mofs;   // 32B aligned
        v8f acc;
        #pragma unroll
        for (int r = 0; r < 8; ++r) acc[r] = e2p[r];

        #pragma unroll
        for (int s = 0; s < 16; ++s) {
            v2f a = *(const v2f*)(arow + (s << 2));   // emb[e][4s+khalf .. +1], 8B aligned
            acc = __builtin_amdgcn_wmma_f32_16x16x4_f32(
                false, a, false, breg[s], (short)0, acc, false, false);
        }

        // branchless running argmax; strict > keeps first max (== jnp.argmin tie rule)
        #pragma unroll
        for (int r = 0; r < 8; ++r) {
            const int   e  = (t << 4) + mofs + r;
            const float v  = acc[r];
            const bool  tk = v > best;
            bestIdx = tk ? e : bestIdx;
            best    = tk ? v : best;
        }
    }

    // merge the two half-waves holding the same z-row (lane ^ 16), branchless
    {
        const float oval = __shfl_xor(best, 16, 32);
        const int   oidx = __shfl_xor(bestIdx, 16, 32);
        const bool  tk   = (oval > best) || (oval == best && oidx < bestIdx);
        bestIdx = tk ? oidx : bestIdx;
        best    = tk ? oval : best;
    }

    if (lane < 16) {
        idx_out[nrow] = bestIdx;
        atomicAdd(&cluster_size[bestIdx], 1.0f);
    }

    // scatter-add sum_encodings: 16 rows, 32 lanes x 2 dims each
    for (int r = 0; r < 16; ++r) {
        const int er = __shfl(bestIdx, r, 32);
        const int nr = nbase + r;
        const float* zr = z + ((nr >> 10) << 16) + (nr & 1023);
        const int c0 = lane << 1;
        atomicAdd(&sum_enc[(er << 6) + c0],     zr[(size_t)c0 * 1024]);
        atomicAdd(&sum_enc[(er << 6) + c0 + 1], zr[(size_t)(c0 + 1) * 1024]);
    }
}

// ---------------- kernel 3: EMA update, codebook normalize, perplexity ---------------
__global__ void vq_ema(const float* __restrict__ ema_cs, const float* __restrict__ ema_w,
                       const float* __restrict__ cs, const float* __restrict__ se,
                       float* __restrict__ new_emb, float* __restrict__ out) {
    __shared__ float red[NE];
    const int e = threadIdx.x;                   // 1024 threads
    const float ncs = DECAYF * ema_cs[e] + OMD * cs[e];

    red[e] = ncs; __syncthreads();
    for (int s = 512; s > 0; s >>= 1) { if (e < s) red[e] += red[e + s]; __syncthreads(); }
    const float n = red[0]; __syncthreads();

    const float p = cs[e] / (float)N_ROWS;
    red[e] = p * __logf(p + 1e-10f); __syncthreads();
    for (int s = 512; s > 0; s >>= 1) { if (e < s) red[e] += red[e + s]; __syncthreads(); }
    if (e == 0) out[PERP_OFF] = __expf(-red[0]);

    const float csn = (ncs + EPSF) / (n + (float)NE * EPSF) * n;
    const float inv = 1.0f / csn;
    const int base = e << 6;
    #pragma unroll 4
    for (int c = 0; c < EDIM; ++c)
        new_emb[base + c] = (DECAYF * ema_w[base + c] + OMD * se[base + c]) * inv;
}

// ---------------- kernel 4: z_q gather, straight-through output (NT), loss ---------------
__global__ void vq_zq(const float* __restrict__ z, const int* __restrict__ idx,
                      const float* __restrict__ new_emb, float* __restrict__ out) {
    __shared__ float red[256];
    const int i  = blockIdx.x * 256 + threadIdx.x;   // direct z-layout index
    const int hw = i & 1023, c = (i >> 10) & 63, b = i >> 16;
    const int n  = (b << 10) | hw;
    const int e  = idx[n];
    const float zv = z[i];
    const float dd = new_emb[(e << 6) + c] - zv;
    __builtin_nontemporal_store(zv + dd, &out[ZQ_OFF + i]);  // z + stop_grad(z_q - z)
    red[threadIdx.x] = dd * dd; __syncthreads();
    for (int s = 128; s > 0; s >>= 1) { if (threadIdx.x < s) red[threadIdx.x] += red[threadIdx.x + s]; __syncthreads(); }
    if (threadIdx.x == 0) atomicAdd(&out[0], BETAF * red[0] / (float)Z_ELEMS);
}

// ---------------- kernel 5: one-hot encodings (134 MB NT stream) + idx as float ---------------
__global__ void vq_onehot(const int* __restrict__ idx, float* __restrict__ out) {
    const int n = blockIdx.x;                        // 32768 blocks
    const int e = idx[n];
    const int c0 = threadIdx.x << 2;                 // 4 codes per thread
    v2f v0 = {0.f, 0.f}, v1 = {0.f, 0.f};
    if (e == c0)          v0.x = 1.0f;
    else if (e == c0 + 1) v0.y = 1.0f;
    else if (e == c0 + 2) v1.x = 1.0f;
    else if (e == c0 + 3) v1.y = 1.0f;
    v2f* base = (v2f*)(out + ME_OFF + (size_t)n * NE + c0);  // 8B aligned
    __builtin_nontemporal_store(v0, base);
    __builtin_nontemporal_store(v1, base + 1);
    if (threadIdx.x == 0) out[IDX_OFF + n] = (float)e;
}

extern "C" void kernel_launch(void* const* d_in, const int* in_sizes, int n_in,
                              void* d_out, int out_size, void* d_ws, size_t ws_size,
                              hipStream_t stream) {
    const float* z      = (const float*)d_in[0];
    const float* emb_w  = (const float*)d_in[1];
    const float* ema_cs = (const float*)d_in[2];
    const float* ema_w  = (const float*)d_in[3];
    float* out = (float*)d_out;

    float* ws_f   = (float*)d_ws;
    float* en2h   = ws_f;                    // 1024
    float* cs     = ws_f + 1024;             // 1024
    float* se     = ws_f + 2048;             // 65536
    float* nemb   = ws_f + 2048 + 65536;     // 65536
    int*   idx    = (int*)(ws_f + 2048 + 2 * 65536);  // 32768  (~664 KB total)

    vq_init  <<<256,   256, 0, stream>>>(cs, se, out);
    vq_enorm <<<4,     256, 0, stream>>>(emb_w, en2h);
    vq_argmin<<<256,   256, 0, stream>>>(z, emb_w, en2h, idx, cs, se);
    vq_ema   <<<1,    1024, 0, stream>>>(ema_cs, ema_w, cs, se, nemb, out);
    vq_zq    <<<8192,  256, 0, stream>>>(z, idx, nemb, out);
    vq_onehot<<<32768, 256, 0, stream>>>(idx, out);
}